// GIN_52913997087426
// MI455X (gfx1250) — compile-verified
//
#include <hip/hip_runtime.h>

typedef __attribute__((ext_vector_type(2))) float v2f;
typedef __attribute__((ext_vector_type(8))) float v8f;

#define HID 96
#define TILE_M 16
#define WAVES_PER_BLOCK 4
#define LDS_STRIDE 100  // 96 + 4 pad: rows r and r+8 land on different LDS banks

// ---------------------------------------------------------------------------
// Scatter-add: agg[dst[e]] += feat[src[e]]  (one thread per edge x float4 chunk)
// feat fits in L2 (19.2 MB), so gathers + atomics resolve at L2, not HBM.
// ---------------------------------------------------------------------------
__global__ void gin_scatter_add(const float* __restrict__ feat,
                                const int* __restrict__ src,
                                const int* __restrict__ dst,
                                float* __restrict__ agg, int nEdge) {
  int gid = blockIdx.x * blockDim.x + threadIdx.x;
  const int CH = HID / 4;  // 24 float4 chunks per row
  int total = nEdge * CH;
  if (gid >= total) return;
  int e = gid / CH;
  int c = gid - e * CH;
  int s = src[e];
  int d = dst[e];
  const float4 v = ((const float4*)(feat + (size_t)s * HID))[c];
  float* p = agg + (size_t)d * HID + c * 4;
  atomicAdd(p + 0, v.x);
  atomicAdd(p + 1, v.y);
  atomicAdd(p + 2, v.z);
  atomicAdd(p + 3, v.w);
}

// ---------------------------------------------------------------------------
// Fused GIN MLP for one layer, fp32 WMMA (V_WMMA_F32_16X16X4_F32):
//   z   = relu((X + AGG) @ Wa + ba)      [16,96] tile per wave
//   out = (relu?)( z @ Wb + bb )         z staged in LDS between the GEMMs
// A-frag (16x4): lane<16 -> A[row][k..k+1], lane>=16 -> A[row][k+2..k+3]
// B-frag (4x16): lane<16 -> B[k..k+1][col], lane>=16 -> B[k+2..k+3][col]
// C/D (16x16):   vgpr r, lane L -> row = r + 8*(L>>4), col = L&15
// ---------------------------------------------------------------------------
template <int N2_TILES, int N2, bool RELU2>
__launch_bounds__(WAVES_PER_BLOCK * 32)
__global__ void gin_mlp(const float* __restrict__ X,
                        const float* __restrict__ AGG,
                        const float* __restrict__ Wa, const float* __restrict__ ba,
                        const float* __restrict__ Wb, const float* __restrict__ bb,
                        float* __restrict__ OUT, int nRows) {
  __shared__ float zbuf[WAVES_PER_BLOCK * TILE_M * LDS_STRIDE];

  const int wave = threadIdx.x >> 5;
  const int lane = threadIdx.x & 31;
  const int nTiles = (nRows + TILE_M - 1) / TILE_M;
  const int tile = blockIdx.x * WAVES_PER_BLOCK + wave;
  if (tile >= nTiles) return;  // wave-uniform: EXEC stays all-ones for WMMA

  const int row0 = tile * TILE_M;
  const int rsel = lane & 15;   // A-row select / B,C column select
  const int khalf = lane >> 4;  // which K-pair this half-wave holds

  int arow = row0 + rsel;
  if (arow >= nRows) arow = nRows - 1;  // clamp reads for a partial last tile
  const float* xrow = X + (size_t)arow * HID;
  const float* grow = AGG + (size_t)arow * HID;

  // -------- GEMM1: (X + AGG) @ Wa, 6 output tiles of 16 cols --------
  v8f acc[6];
#pragma unroll
  for (int n = 0; n < 6; ++n)
#pragma unroll
    for (int r = 0; r < 8; ++r) acc[n][r] = 0.0f;

#pragma unroll
  for (int k = 0; k < HID; k += 4) {
    const int kc = k + 2 * khalf;
    v2f xa = *(const v2f*)(xrow + kc);
    v2f ga = *(const v2f*)(grow + kc);
    v2f a;
    a[0] = xa[0] + ga[0];  // fuse the GIN residual add into the A fragment
    a[1] = xa[1] + ga[1];
#pragma unroll
    for (int n = 0; n < 6; ++n) {
      const int col = n * 16 + rsel;
      v2f b;
      b[0] = Wa[(size_t)kc * HID + col];
      b[1] = Wa[(size_t)(kc + 1) * HID + col];
      acc[n] = __builtin_amdgcn_wmma_f32_16x16x4_f32(
          false, a, false, b, (short)0, acc[n], false, false);
    }
  }

  // bias + ReLU, stage z tile in LDS (padded stride -> bank-conflict free)
  float* zb = zbuf + wave * (TILE_M * LDS_STRIDE);
#pragma unroll
  for (int n = 0; n < 6; ++n) {
    const int col = n * 16 + rsel;
    const float bias = ba[col];
#pragma unroll
    for (int r = 0; r < 8; ++r) {
      float v = acc[n][r] + bias;
      v = fmaxf(v, 0.0f);
      zb[(r + 8 * khalf) * LDS_STRIDE + col] = v;
    }
  }
  asm volatile("s_wait_dscnt 0" ::: "memory");  // own-wave LDS store->load fence

  // -------- GEMM2: z @ Wb, N2 output cols (padded to N2_TILES*16) --------
  v8f acc2[N2_TILES];
#pragma unroll
  for (int n = 0; n < N2_TILES; ++n)
#pragma unroll
    for (int r = 0; r < 8; ++r) acc2[n][r] = 0.0f;

#pragma unroll
  for (int k = 0; k < HID; k += 4) {
    const int kc = k + 2 * khalf;
    v2f a = *(const v2f*)(zb + rsel * LDS_STRIDE + kc);  // 8B-aligned, no conflicts
#pragma unroll
    for (int n = 0; n < N2_TILES; ++n) {
      const int col = n * 16 + rsel;
      v2f b;
      if ((N2 % 16 == 0) || (col < N2)) {
        b[0] = Wb[(size_t)kc * N2 + col];
        b[1] = Wb[(size_t)(kc + 1) * N2 + col];
      } else {
        b[0] = 0.0f;
        b[1] = 0.0f;
      }
      acc2[n] = __builtin_amdgcn_wmma_f32_16x16x4_f32(
          false, a, false, b, (short)0, acc2[n], false, false);
    }
  }

  // bias (+ optional ReLU) and store
#pragma unroll
  for (int n = 0; n < N2_TILES; ++n) {
    const int col = n * 16 + rsel;
    if ((N2 % 16 == 0) || (col < N2)) {
      const float bias = bb[col];
#pragma unroll
      for (int r = 0; r < 8; ++r) {
        const int row = row0 + r + 8 * khalf;
        if (row < nRows) {
          float v = acc2[n][r] + bias;
          if (RELU2) v = fmaxf(v, 0.0f);
          OUT[(size_t)row * N2 + col] = v;
        }
      }
    }
  }
}

// ---------------------------------------------------------------------------
extern "C" void kernel_launch(void* const* d_in, const int* in_sizes, int n_in,
                              void* d_out, int out_size, void* d_ws, size_t ws_size,
                              hipStream_t stream) {
  const float* x   = (const float*)d_in[0];
  const int*  edge = (const int*)d_in[1];  // [2,E]; JAX (x64 off) -> int32
  const float* W1a = (const float*)d_in[2];
  const float* b1a = (const float*)d_in[3];
  const float* W1b = (const float*)d_in[4];
  const float* b1b = (const float*)d_in[5];
  const float* W2a = (const float*)d_in[6];
  const float* b2a = (const float*)d_in[7];
  const float* W2b = (const float*)d_in[8];
  const float* b2b = (const float*)d_in[9];
  float* out = (float*)d_out;

  const int nNodes = in_sizes[0] / HID;  // 50000
  const int nEdge  = in_sizes[1] / 2;    // 800000
  const int* src = edge;
  const int* dst = edge + nEdge;

  float* agg = (float*)d_ws;                  // [N,96] aggregation buffer
  float* h1  = agg + (size_t)nNodes * HID;    // [N,96] layer-1 output

  const int nTiles = (nNodes + TILE_M - 1) / TILE_M;
  dim3 mlpGrid((nTiles + WAVES_PER_BLOCK - 1) / WAVES_PER_BLOCK);
  dim3 mlpBlock(WAVES_PER_BLOCK * 32);

  const int scThreads = nEdge * (HID / 4);
  dim3 scGrid((scThreads + 255) / 256);
  dim3 scBlock(256);

  const size_t aggBytes = (size_t)nNodes * HID * sizeof(float);

  // Layer 1: agg = scatter(x); h1 = relu(MLP1(x + agg))
  hipMemsetAsync(agg, 0, aggBytes, stream);
  gin_scatter_add<<<scGrid, scBlock, 0, stream>>>(x, src, dst, agg, nEdge);
  gin_mlp<6, 96, true><<<mlpGrid, mlpBlock, 0, stream>>>(
      x, agg, W1a, b1a, W1b, b1b, h1, nNodes);

  // Layer 2: agg = scatter(h1); out = MLP2(h1 + agg)   [N,40]
  hipMemsetAsync(agg, 0, aggBytes, stream);
  gin_scatter_add<<<scGrid, scBlock, 0, stream>>>(h1, src, dst, agg, nEdge);
  gin_mlp<3, 40, false><<<mlpGrid, mlpBlock, 0, stream>>>(
      h1, agg, W2a, b2a, W2b, b2b, out, nNodes);
}